// VoxelPointsSampler_62448824484384
// MI455X (gfx1250) — compile-verified
//
#include <hip/hip_runtime.h>
#include <hip/hip_bf16.h>
#include <math.h>

#define VOXEL_SIZE 0.4f
#define GAMMA      1.1f
#define FDIM       5

typedef float v2f __attribute__((ext_vector_type(2)));
typedef float v8f __attribute__((ext_vector_type(8)));

// ---------------------------------------------------------------------------
// k0: per-box precompute.
//   boxPrep[r] = {bx, by, bx^2+by^2, r^2}            (point space)
//   voxPrep[r] = {qx, qy, rvox^2, 0}                 (voxel-grid space)
// ---------------------------------------------------------------------------
__global__ void k0_box_prep(const float* __restrict__ boxes,
                            const float* __restrict__ pc_start,
                            float4* __restrict__ boxPrep,
                            float4* __restrict__ voxPrep,
                            int R) {
    int r = blockIdx.x * blockDim.x + threadIdx.x;
    if (r >= R) return;
    float bx = boxes[r * 7 + 0], by = boxes[r * 7 + 1];
    float dx = boxes[r * 7 + 3], dy = boxes[r * 7 + 4];
    float rad  = sqrtf(dx * dx * 0.25f + dy * dy * 0.25f) * GAMMA;
    float qx   = (bx - pc_start[0]) * (1.0f / VOXEL_SIZE);
    float qy   = (by - pc_start[1]) * (1.0f / VOXEL_SIZE);
    float rvox = rad * (1.0f / VOXEL_SIZE);
    boxPrep[r] = make_float4(bx, by, bx * bx + by * by, rad * rad);
    voxPrep[r] = make_float4(qx, qy, rvox * rvox, 0.0f);
}

// ---------------------------------------------------------------------------
// k1: voxel-level radius search + packed per-point validity bits.
// Each thread handles 2 voxels (K=16 points each) -> one u32 validBits word.
// Also emits voxel_mask as floats into d_out.
// ---------------------------------------------------------------------------
__global__ void k1_voxel_mask(const int* __restrict__ coords,
                              const int* __restrict__ num_points,
                              const float4* __restrict__ voxPrep,
                              unsigned* __restrict__ validBits,
                              float* __restrict__ outVoxMask,
                              int V, int R) {
    __shared__ float4 sBox[256];
    for (int i = threadIdx.x; i < R; i += blockDim.x) sBox[i] = voxPrep[i];
    __syncthreads();

    int t  = blockIdx.x * blockDim.x + threadIdx.x;   // word index
    int v0 = t * 2;
    if (v0 >= V) return;

    unsigned word = 0u;
    #pragma unroll
    for (int j = 0; j < 2; ++j) {
        int v = v0 + j;
        unsigned bits = 0u;
        if (v < V) {
            float cx = (float)coords[v * 2 + 0];
            float cy = (float)coords[v * 2 + 1];
            int any = 0;
            for (int r = 0; r < R; ++r) {
                float4 b = sBox[r];
                float ddx = b.x - cx, ddy = b.y - cy;
                any |= (ddx * ddx + ddy * ddy < b.z) ? 1 : 0;
            }
            int np = num_points[v];
            if (np < 0)  np = 0;
            if (np > 16) np = 16;
            unsigned kb = (np >= 32) ? 0xFFFFFFFFu : ((1u << np) - 1u);
            bits = any ? (kb & 0xFFFFu) : 0u;
            outVoxMask[v] = any ? 1.0f : 0.0f;
        }
        word |= bits << (16 * j);
    }
    validBits[t] = word;
}

// ---------------------------------------------------------------------------
// k2: WMMA pairwise-distance mask.
//   dist^2(m,n) = A[m] . B[n]  with  A[m] = [bx, by, |b|^2, 1],
//                                    B[n] = [-2px, -2py, 1, |p|^2]^T
// One wave handles a 16-box tile x 32-point chunk (2x v_wmma_f32_16x16x4_f32),
// compares D <= r^2, ANDs validity, ballots the predicates into bitmask words.
// ---------------------------------------------------------------------------
__global__ void k2_point_mask(const float* __restrict__ voxels,      // [N, F]
                              const float4* __restrict__ boxPrep,
                              const unsigned* __restrict__ validBits,
                              unsigned* __restrict__ maskWords,      // [R, W]
                              int R, int W) {
    int lane = threadIdx.x & 31;
    int wid  = (int)((blockIdx.x * blockDim.x + threadIdx.x) >> 5);
    int BT   = R >> 4;
    if (wid >= BT * W) return;               // wave-uniform: EXEC stays all-ones
    int bt    = wid % BT;
    int chunk = wid / BT;                    // mask word index (32 points)

    // A matrix: lanes 0-15 -> {bx, by} (K=0,1); lanes 16-31 -> {|b|^2, 1} (K=2,3)
    int mrow  = (bt << 4) + (lane & 15);
    float4 bp = boxPrep[mrow];
    v2f a;
    a.x = (lane < 16) ? bp.x : bp.z;
    a.y = (lane < 16) ? bp.y : 1.0f;

    // radii^2 for the 8 box rows this lane's half of the wave covers
    float r2arr[8];
    int halfOff = (lane >> 4) << 3;          // 0 or 8
    #pragma unroll
    for (int j = 0; j < 8; ++j)
        r2arr[j] = boxPrep[(bt << 4) + j + halfOff].w;

    unsigned vw = validBits[chunk];

    unsigned bal[2][8];
    int n0 = chunk << 5;
    #pragma unroll
    for (int t = 0; t < 2; ++t) {
        int n = n0 + (t << 4) + (lane & 15);
        float px = voxels[(size_t)n * FDIM + 0];
        float py = voxels[(size_t)n * FDIM + 1];
        v2f b;
        b.x = (lane < 16) ? (-2.0f * px) : 1.0f;
        b.y = (lane < 16) ? (-2.0f * py) : (px * px + py * py);

        v8f c = {0.f, 0.f, 0.f, 0.f, 0.f, 0.f, 0.f, 0.f};
        c = __builtin_amdgcn_wmma_f32_16x16x4_f32(false, a, false, b,
                                                  (short)0, c, false, false);

        unsigned myvalid = (vw >> ((t << 4) + (lane & 15))) & 1u;
        #pragma unroll
        for (int j = 0; j < 8; ++j) {
            bool pred = (c[j] <= r2arr[j]) && (myvalid != 0u);
            bal[t][j] = __builtin_amdgcn_ballot_w32(pred);
        }
    }

    // Ballot results are wave-uniform; lane 0 assembles & stores 16 words.
    if (lane == 0) {
        #pragma unroll
        for (int j = 0; j < 8; ++j) {
            unsigned lo0 = bal[0][j] & 0xFFFFu, hi0 = bal[0][j] >> 16;
            unsigned lo1 = bal[1][j] & 0xFFFFu, hi1 = bal[1][j] >> 16;
            unsigned wlow  = lo0 | (lo1 << 16);                // box bt*16 + j
            unsigned whigh = hi0 | (hi1 << 16);                // box bt*16 + j + 8
            maskWords[(size_t)((bt << 4) + j)     * W + chunk] = wlow;
            maskWords[(size_t)((bt << 4) + j + 8) * W + chunk] = whigh;
        }
    }
}

// ---------------------------------------------------------------------------
// k3: ordered compaction == top_k on a 0/1 mask with lower-index tie-break.
// One wave per ROI: popcount + shfl prefix-scan over mask words, emit the
// first S set indices (gather 5 features each), zero-fill the tail.
// ---------------------------------------------------------------------------
__global__ void k3_gather(const unsigned* __restrict__ maskWords,   // [R, W]
                          const float* __restrict__ voxels,         // [N, F]
                          float* __restrict__ outKP,                // [R, S, F]
                          float* __restrict__ outMask,              // [R, S]
                          int W, int S) {
    int r    = blockIdx.x;
    int lane = threadIdx.x;                  // 32 threads per block
    const unsigned* row = maskWords + (size_t)r * W;
    float* kp = outKP   + (size_t)r * S * FDIM;
    float* mk = outMask + (size_t)r * S;

    int total = 0;
    for (int w0 = 0; w0 < W && total < S; w0 += 32) {
        int w = w0 + lane;
        unsigned word = (w < W) ? row[w] : 0u;
        int cnt  = __popc(word);
        int incl = cnt;
        #pragma unroll
        for (int d = 1; d < 32; d <<= 1) {
            int t = __shfl_up(incl, d, 32);
            if (lane >= d) incl += t;
        }
        int excl = total + (incl - cnt);
        int rank = 0;
        unsigned tmp = word;
        while (tmp) {
            int b = __ffs((int)tmp) - 1;
            tmp &= tmp - 1u;
            int pos = excl + rank;
            ++rank;
            if (pos < S) {
                int n = (w << 5) + b;
                #pragma unroll
                for (int f = 0; f < FDIM; ++f)
                    kp[pos * FDIM + f] = voxels[(size_t)n * FDIM + f];
                mk[pos] = 1.0f;
            }
        }
        total += __shfl(incl, 31, 32);
    }
    if (total > S) total = S;
    for (int pos = total + lane; pos < S; pos += 32) {
        #pragma unroll
        for (int f = 0; f < FDIM; ++f) kp[pos * FDIM + f] = 0.0f;
        mk[pos] = 0.0f;
    }
}

// ---------------------------------------------------------------------------
extern "C" void kernel_launch(void* const* d_in, const int* in_sizes, int n_in,
                              void* d_out, int out_size, void* d_ws, size_t ws_size,
                              hipStream_t stream) {
    const float* voxels     = (const float*)d_in[0];
    const float* boxes      = (const float*)d_in[1];
    const float* pc_start   = (const float*)d_in[2];
    const int*   coords     = (const int*)d_in[3];
    const int*   num_points = (const int*)d_in[4];
    (void)coords; (void)n_in;

    // Derive shapes from sizes (reference: V=20000, K=16, F=5, R=256, S=128).
    int V = in_sizes[4];
    int R = in_sizes[1] / 7;
    int F = FDIM;
    int K = in_sizes[0] / (V * F);           // == 16
    int N = V * K;
    int W = N / 32;                          // == V/2 when K==16
    int S = (out_size - V) / (R * (F + 1));

    float* outKP   = (float*)d_out;
    float* outMask = outKP + (size_t)R * S * F;
    float* outVox  = outMask + (size_t)R * S;

    // Workspace layout (256B-aligned sections).
    char* ws = (char*)d_ws;
    size_t off = 0;
    float4* boxPrep = (float4*)(ws + off); off += (size_t)R * 16; off = (off + 255) & ~(size_t)255;
    float4* voxPrep = (float4*)(ws + off); off += (size_t)R * 16; off = (off + 255) & ~(size_t)255;
    unsigned* validBits = (unsigned*)(ws + off); off += (size_t)W * 4; off = (off + 255) & ~(size_t)255;
    unsigned* maskWords = (unsigned*)(ws + off);
    (void)ws_size;

    // k0: box precompute
    k0_box_prep<<<(R + 255) / 256, 256, 0, stream>>>(boxes, pc_start, boxPrep, voxPrep, R);

    // k1: voxel mask + validity bits
    int pairs = (V + 1) / 2;
    k1_voxel_mask<<<(pairs + 255) / 256, 256, 0, stream>>>(
        coords, num_points, voxPrep, validBits, outVox, V, R);

    // k2: WMMA pairwise-distance bitmask
    long totalWaves   = (long)(R / 16) * W;
    long totalThreads = totalWaves * 32;
    k2_point_mask<<<(int)((totalThreads + 255) / 256), 256, 0, stream>>>(
        voxels, boxPrep, validBits, maskWords, R, W);

    // k3: ordered compaction + feature gather
    k3_gather<<<R, 32, 0, stream>>>(maskWords, voxels, outKP, outMask, W, S);
}